// FourierELU_88227218195137
// MI455X (gfx1250) — compile-verified
//
#include <hip/hip_runtime.h>
#include <hip/hip_bf16.h>

typedef __attribute__((ext_vector_type(2))) float v2f;
typedef __attribute__((ext_vector_type(8))) float v8f;

#define GROUP 8
#define PIX_PER_BC 4096                 // 64*64 pixels under each (b,c)
#define BC_STRIDE (GROUP * PIX_PER_BC)  // 32768 elements per (b,c) slice
#define TOTAL_PIX (32 * 64 * 64 * 64)   // 8,388,608 independent signals
#define TILES_PER_WAVE 8                // 8 tiles * 16 pixels = 128 pixels per wave

// lane L receives value from lane L^16 (group-of-32 swizzle, xor=0x10, and=0x1f).
// Must be executed with full EXEC: disabled source lanes would return 0.
__device__ __forceinline__ float swap16(float v) {
    return __int_as_float(__builtin_amdgcn_ds_swizzle(__float_as_int(v), 0x401F));
}

// out = D * elu(U * x) per pixel, fused, one streaming pass over HBM.
// U[m][n] = f((m-2n)&15), D[n][m] = g((m-2n)&15)  (closed-form DFT algebra).
__global__ __launch_bounds__(256) void fourier_elu_wmma(const float* __restrict__ x,
                                                        float* __restrict__ out) {
    __shared__ float sF[16];
    __shared__ float sG[16];

    const int tid = threadIdx.x;
    if (tid < 32) {
        int d = tid & 15;
        float c1 = cospif((float)d * 0.125f);   // cos(pi*d/8)
        float c2 = cospif((float)d * 0.250f);   // cos(pi*2d/8)
        float c3 = cospif((float)d * 0.375f);   // cos(pi*3d/8)
        float c4 = cospif((float)d * 0.500f);   // cos(pi*4d/8)
        if (tid < 16)
            sF[d] = (1.0f + 2.0f * (c1 + c2 + c3 + c4)) * (1.0f / 16.0f);
        else
            sG[d] = (1.0f + 2.0f * (c1 + c2 + c3) + c4) * 0.125f;
    }
    __syncthreads();

    const int lane = tid & 31;
    const int wave = tid >> 5;
    const int row  = lane & 15;   // M (A) or N (B/C) index
    const int half = lane >> 4;   // 0 = lanes 0-15, 1 = lanes 16-31
    const int cb   = half << 1;   // K sub-offset inside a K=4 chunk: 0 or 2

    // ---- A matrices (lane-resident constants, computed ONCE per wave) ----
    // A 16x4 f32 layout (ISA-documented): V0 = {K=0 | K=2}, V1 = {K=1 | K=3}.
    v2f a0, a1;                                    // U columns 0-3 and 4-7
    a0.x = sF[(row - 2 * (cb + 0)) & 15];
    a0.y = sF[(row - 2 * (cb + 1)) & 15];
    a1.x = sF[(row - 2 * (cb + 4)) & 15];
    a1.y = sF[(row - 2 * (cb + 5)) & 15];

    v2f da[4];                                     // D padded to 16x16 (rows 8-15 zero)
#pragma unroll
    for (int j = 0; j < 4; ++j) {
        int k0 = 4 * j + cb;
        da[j].x = (row < 8) ? sG[(k0 + 0 - 2 * row) & 15] : 0.0f;
        da[j].y = (row < 8) ? sG[(k0 + 1 - 2 * row) & 15] : 0.0f;
    }

    // ---- this wave's 128-pixel run (128-aligned => never straddles a bc slice) ----
    const long run    = ((long)blockIdx.x * 8 + wave) * (16L * TILES_PER_WAVE);
    const long bc     = run >> 12;                 // / 4096
    const long within = run & 4095;
    const long base   = bc * (long)BC_STRIDE + within + row;

    const float* px = x   + base + (long)cb   * PIX_PER_BC;  // lanes hi pre-offset by 2 rows
    float*       po = out + base + (long)half * PIX_PER_BC;  // store k = 2r + half

#pragma unroll 2
    for (int t = 0; t < TILES_PER_WAVE; ++t) {
        const int o = t * 16;

        // ---- load B (8x16): lanes lo get K={0,1,4,5}, hi K={2,3,6,7} ----
        v2f b0, b1;
        b0.x = __builtin_nontemporal_load(px + o);
        b0.y = __builtin_nontemporal_load(px + o + 1 * PIX_PER_BC);
        b1.x = __builtin_nontemporal_load(px + o + 4 * PIX_PER_BC);
        b1.y = __builtin_nontemporal_load(px + o + 5 * PIX_PER_BC);

        // ---- GEMM 1: Y(16x16) = U(16x8) * X(8x16), two chained K=4 WMMAs ----
        v8f c = {};
        c = __builtin_amdgcn_wmma_f32_16x16x4_f32(false, a0, false, b0, (short)0, c, false, false);
        c = __builtin_amdgcn_wmma_f32_16x16x4_f32(false, a1, false, b1, (short)0, c, false, false);

        // ---- ELU (alpha=1), branch-free via hardware exp ----
#pragma unroll
        for (int r = 0; r < 8; ++r) {
            float v = c[r];
            float e = __expf(v) - 1.0f;
            c[r] = (v > 0.0f) ? v : e;
        }

        // ---- C/D layout -> B layout: swizzles first (FULL exec), then pure selects ----
        float sw[8];
#pragma unroll
        for (int r = 0; r < 8; ++r) sw[r] = swap16(c[r]);

        v2f bb[4];
        bb[0].x = half ? sw[2] : c[0];   bb[0].y = half ? sw[3] : c[1];   // Y rows 0-3
        bb[1].x = half ? sw[6] : c[4];   bb[1].y = half ? sw[7] : c[5];   // Y rows 4-7
        bb[2].x = half ? c[2]  : sw[0];  bb[2].y = half ? c[3]  : sw[1];  // Y rows 8-11
        bb[3].x = half ? c[6]  : sw[4];  bb[3].y = half ? c[7]  : sw[5];  // Y rows 12-15

        // ---- GEMM 2: out(16x16, rows 0-7 valid) = Dp(16x16) * Y(16x16) ----
        v8f dacc = {};
#pragma unroll
        for (int j = 0; j < 4; ++j)
            dacc = __builtin_amdgcn_wmma_f32_16x16x4_f32(false, da[j], false, bb[j],
                                                         (short)0, dacc, false, false);

        // ---- pack rows 0-7 into 4 full-wave NT stores (swizzles at full exec) ----
        float sd[4];
#pragma unroll
        for (int r = 0; r < 4; ++r) sd[r] = swap16(dacc[2 * r + 1]);
#pragma unroll
        for (int r = 0; r < 4; ++r) {
            float val = half ? sd[r] : dacc[2 * r];
            __builtin_nontemporal_store(val, po + o + (long)(2 * r) * PIX_PER_BC);
        }
    }
}

extern "C" void kernel_launch(void* const* d_in, const int* in_sizes, int n_in,
                              void* d_out, int out_size, void* d_ws, size_t ws_size,
                              hipStream_t stream) {
    (void)in_sizes; (void)n_in; (void)out_size; (void)d_ws; (void)ws_size;
    const float* x = (const float*)d_in[0];
    float* out     = (float*)d_out;
    // 8,388,608 pixels / (128 per wave * 8 waves per block) = 8192 blocks
    dim3 grid(TOTAL_PIX / (128 * 8));
    fourier_elu_wmma<<<grid, 256, 0, stream>>>(x, out);
}